// SparseGrid_1726576857331
// MI455X (gfx1250) — compile-verified
//
#include <hip/hip_runtime.h>
#include <stdint.h>

// Grid is fixed 128^3 by the reference (RESO), 28 output channels (1 sigma + 27 sh).
#define RDIM 128
#define CH   28
#define PPW  4                 // points per wave (sequential, unrolled)
#define NWAVES 8               // waves per block
#define PPB  (PPW * NWAVES)    // 32 points per block
#define NTHREADS 256

__global__ __launch_bounds__(NTHREADS)
void trilerp_gather_kernel(const float* __restrict__ points,
                           const float* __restrict__ density,
                           const float* __restrict__ sh,
                           const int*   __restrict__ links,
                           float* __restrict__ out,
                           int n)
{
    __shared__ float s_pts[PPB * 3];   // 384 B: this block's point coords

    const int tid = threadIdx.x;
    const int p0  = blockIdx.x * PPB;

    // ---- Stage point coords global->LDS with the gfx1250 async path ----
    // Per-lane async copy (EXEC-honored, ASYNCcnt-tracked). LDS operand is the
    // workgroup-relative LDS byte offset == low 32 bits of the generic address
    // of a __shared__ object.
    if (tid < PPB * 3) {
        const int g = p0 * 3 + tid;
        if (g < n * 3) {
            uint32_t lds_off = (uint32_t)(uintptr_t)(&s_pts[tid]);
            uint32_t goff    = (uint32_t)g * 4u;
            asm volatile("global_load_async_to_lds_b32 %0, %1, %2 offset:0"
                         :: "v"(lds_off), "v"(goff), "s"(points)
                         : "memory");
        } else {
            s_pts[tid] = 0.0f;
        }
    }
    asm volatile("s_wait_asynccnt 0x0" ::: "memory");
    __syncthreads();

    const int  lane   = tid & 31;          // wave32
    const int  wave   = tid >> 5;
    const bool ch_act = (lane < CH);

    // lane 0 gathers density (row stride 1); lanes 1..27 gather sh channel lane-1
    // (row stride 27). One per-lane base pointer -> a single global_load_b32
    // per corner covers both tables, 108B contiguous per voxel row.
    const float* gbase = (lane == 0) ? density : (sh + (lane - 1));
    const int    gstep = (lane == 0) ? 1 : 27;

    #pragma unroll
    for (int i = 0; i < PPW; ++i) {
        const int p = p0 + wave * PPW + i;
        if (p < n) {
            const int sp = (wave * PPW + i) * 3;
            // p_grid = 128*(0.5 + 0.5*x) - 0.5, clipped to [0,127]
            float fx = fminf(fmaxf((float)RDIM * (0.5f + 0.5f * s_pts[sp + 0]) - 0.5f, 0.0f), (float)(RDIM - 1));
            float fy = fminf(fmaxf((float)RDIM * (0.5f + 0.5f * s_pts[sp + 1]) - 0.5f, 0.0f), (float)(RDIM - 1));
            float fz = fminf(fmaxf((float)RDIM * (0.5f + 0.5f * s_pts[sp + 2]) - 0.5f, 0.0f), (float)(RDIM - 1));
            int ix = (int)fx; if (ix > RDIM - 2) ix = RDIM - 2;
            int iy = (int)fy; if (iy > RDIM - 2) iy = RDIM - 2;
            int iz = (int)fz; if (iz > RDIM - 2) iz = RDIM - 2;
            const float wbx = fx - (float)ix, wax = 1.0f - wbx;
            const float wby = fy - (float)iy, way = 1.0f - wby;
            const float wbz = fz - (float)iz, waz = 1.0f - wbz;

            float acc = 0.0f;
            #pragma unroll
            for (int dx = 0; dx < 2; ++dx) {
                const float wx = dx ? wbx : wax;
                const int   bx = (ix + dx) * RDIM;
                #pragma unroll
                for (int dy = 0; dy < 2; ++dy) {
                    const float wxy = wx * (dy ? wby : way);
                    const int   bxy = (bx + iy + dy) * RDIM;
                    #pragma unroll
                    for (int dz = 0; dz < 2; ++dz) {
                        const float w  = wxy * (dz ? wbz : waz);
                        const int   lk = links[bxy + iz + dz];   // wave-uniform broadcast load
                        const float we = (lk >= 0) ? w : 0.0f;
                        const int safe = (lk >= 0) ? lk : 0;
                        float v = 0.0f;
                        if (ch_act) v = gbase[(size_t)safe * gstep]; // temporal: keep in L2
                        acc = fmaf(we, v, acc);
                    }
                }
            }
            if (ch_act) {
                // Streaming output (TH=NT): don't evict the 226MB gather tables
                // from the 192MB L2 with write-once data.
                __builtin_nontemporal_store(acc, out + (size_t)p * CH + lane);
            }
        }
    }
}

extern "C" void kernel_launch(void* const* d_in, const int* in_sizes, int n_in,
                              void* d_out, int out_size, void* d_ws, size_t ws_size,
                              hipStream_t stream)
{
    const float* points  = (const float*)d_in[0];  // (N,3)
    const float* density = (const float*)d_in[1];  // (128^3, 1)
    const float* sh      = (const float*)d_in[2];  // (128^3, 27)
    const int*   links   = (const int*)d_in[3];    // (128,128,128)
    float*       out     = (float*)d_out;          // (N, 28)

    const int n = in_sizes[0] / 3;
    const int nblocks = (n + PPB - 1) / PPB;
    hipLaunchKernelGGL(trilerp_gather_kernel, dim3(nblocks), dim3(NTHREADS), 0, stream,
                       points, density, sh, links, out, n);
}